// FourierWeightLinear_3899830305372
// MI455X (gfx1250) — compile-verified
//
#include <hip/hip_runtime.h>
#include <math.h>

typedef __attribute__((ext_vector_type(2))) float v2f;
typedef __attribute__((ext_vector_type(8))) float v8f;
typedef __attribute__((ext_vector_type(4))) unsigned int u32x4;
typedef __attribute__((ext_vector_type(8))) int i32x8;
typedef __attribute__((ext_vector_type(4))) int i32x4;

#define IN_DIM   1024
#define OUT_DIM  1024
#define TOTAL_K  96
#define BATCH    4096
// SCALE = sqrt(2/(1024+1024)) = sqrt(1/1024) = 1/32 exactly
#define W_SCALE  0.03125f

// ---------------------------------------------------------------------------
// Phase 1: synthesize W[i][j] (VALU/transcendental bound)
// ---------------------------------------------------------------------------
__global__ void __launch_bounds__(256)
gen_w_kernel(const float* __restrict__ omega,   // (96,3) row-major
             const float* __restrict__ phi,     // (96,)
             const float* __restrict__ alpha,   // (192,)  [0:96]=sin, [96:192]=cos
             float* __restrict__ W)             // (1024,1024) row-major
{
    __shared__ float s_om[TOTAL_K * 3];
    __shared__ float s_phi[TOTAL_K];
    __shared__ float s_as[TOTAL_K];
    __shared__ float s_ac[TOTAL_K];

    const int t = threadIdx.x;
    if (t < TOTAL_K) {
        s_phi[t] = phi[t];
        s_as[t]  = alpha[t];
        s_ac[t]  = alpha[TOTAL_K + t];
    }
    for (int i = t; i < TOTAL_K * 3; i += 256) s_om[i] = omega[i];
    __syncthreads();

    const int gid = blockIdx.x * 256 + t;        // 0 .. 1M-1
    const int i = gid >> 10;                     // row in IN_DIM
    const int j = gid & (OUT_DIM - 1);           // col in OUT_DIM

    const float ii = (float)(i + 1) * (1.0f / (float)(IN_DIM + 1));
    const float jj = (float)(j + 1) * (1.0f / (float)(OUT_DIM + 1));
    const float ll = 0.5f;                       // (LAYER_IDX+1)/(NUM_LAYERS+1)

    float acc = 0.0f;
#pragma unroll 8
    for (int k = 0; k < TOTAL_K; ++k) {
        float arg = fmaf(ii, s_om[3 * k + 0],
                    fmaf(jj, s_om[3 * k + 1],
                    fmaf(ll, s_om[3 * k + 2], s_phi[k])));
        acc = fmaf(sinf(arg), s_as[k], acc);
        acc = fmaf(cosf(arg), s_ac[k], acc);
    }
    W[gid] = acc * W_SCALE;
}

// ---------------------------------------------------------------------------
// Phase 2: out = x @ W + bias, fp32 WMMA + TDM double-buffered LDS staging.
// Block = 256 threads = 8 wave32s.  Block tile 32(M) x 64(N), BK = 32.
// Wave 0 posts TENSOR_LOAD_TO_LDS descriptors (2-D tiles, HW row padding),
// tracked on TENSORcnt; compute overlaps the next chunk's DMA.
// ---------------------------------------------------------------------------
#define BM 32
#define BN 64
#define BK 32
#define AS_STRIDE 36   // 32 + 4 DWORD pad (TDM pad_amount)
#define BS_STRIDE 68   // 64 + 4 DWORD pad

// Build a 2-D tile D# (ISA 8.3/8.4) and post TENSOR_LOAD_TO_LDS.
// pad: interval = 2^(icode+1) DWORDs (= tile row width), amount = acode+1 DWORDs.
__device__ __forceinline__ void tdm_load_tile_2d(
    const float* __restrict__ gptr,  // global addr of tile origin
    unsigned lds_byte_off,           // LDS byte offset of tile dest
    int tensor_w, int tensor_h,      // full tensor dims (elements)
    int tile_w, int tile_h,          // tile dims (elements)
    int row_stride,                  // tensor dim0 stride (elements)
    int icode, int acode)            // pad interval / amount codes
{
    unsigned long long ga = (unsigned long long)gptr;
    u32x4 g0;
    g0[0] = 1u;                                            // count=1, user D#
    g0[1] = lds_byte_off;                                  // lds_addr
    g0[2] = (unsigned)(ga & 0xFFFFFFFFu);                  // global_addr[31:0]
    g0[3] = (unsigned)((ga >> 32) & 0x01FFFFFFu)           // global_addr[56:32]
          | (2u << 30);                                    // type = 2 ("image")
    i32x8 g1;
    g1[0] = (2 << 16)            // data_size = 4 bytes
          | (1 << 20)            // pad_enable
          | (icode << 22)        // pad_interval
          | (acode << 25);       // pad_amount
    g1[1] = (tensor_w & 0xFFFF) << 16;                            // dim0[15:0]
    g1[2] = ((tensor_w >> 16) & 0xFFFF) | ((tensor_h & 0xFFFF) << 16);
    g1[3] = ((tensor_h >> 16) & 0xFFFF) | (tile_w << 16);         // tile_dim0
    g1[4] = tile_h;                                               // tile_dim1 (tile_dim2=0)
    g1[5] = row_stride;                                           // dim0_stride[31:0]
    g1[6] = 0;
    g1[7] = 0;
    i32x4 g2 = {0, 0, 0, 0};     // 2-D tensor: groups 2/3 unused
    i32x4 g3 = {0, 0, 0, 0};
    i32x8 g4 = {0, 0, 0, 0, 0, 0, 0, 0};  // extra group (6-arg toolchain form)
    __builtin_amdgcn_tensor_load_to_lds(g0, g1, g2, g3, g4, 0);
}

__device__ __forceinline__ unsigned lds_off_of(const void* p) {
    return (unsigned)(unsigned long long)(uintptr_t)p;
}

__global__ void __launch_bounds__(256)
wmma_gemm_kernel(const float* __restrict__ X,    // (4096,1024)
                 const float* __restrict__ W,    // (1024,1024)
                 const float* __restrict__ bias, // (1024,)
                 float* __restrict__ Out)        // (4096,1024)
{
    __shared__ float As[2][BM * AS_STRIDE];   // [buf][row][k]   (padded rows)
    __shared__ float Bs[2][BK * BS_STRIDE];   // [buf][k][col]   (padded rows)

    const int t    = threadIdx.x;
    const int lane = t & 31;
    const int wave = t >> 5;               // 0..7
    const int waveM = wave >> 2;           // 0..1
    const int waveN = wave & 3;            // 0..3
    const int half = lane >> 4;            // 0: lanes 0-15, 1: lanes 16-31
    const int l    = lane & 15;

    const int m0 = blockIdx.y * BM;
    const int n0 = blockIdx.x * BN;

    const int NCHUNK = IN_DIM / BK;        // 32

    // Prologue: post chunk 0 tiles into buffer 0 (wave 0 only; scalar branch).
    if (wave == 0) {
        tdm_load_tile_2d(X + (size_t)m0 * IN_DIM, lds_off_of(&As[0][0]),
                         IN_DIM, BATCH, BK, BM, IN_DIM, /*32 dw*/4, /*4 dw*/3);
        tdm_load_tile_2d(W + n0, lds_off_of(&Bs[0][0]),
                         OUT_DIM, IN_DIM, BN, BK, OUT_DIM, /*64 dw*/5, /*4 dw*/3);
    }

    v8f acc = {};
    const int arow = waveM * 16 + l;
    const int bcol = waveN * 16 + l;

    for (int c = 0; c < NCHUNK; ++c) {
        const int buf = c & 1;
        if (wave == 0) {
            if (c + 1 < NCHUNK) {
                const int k1 = (c + 1) * BK;
                tdm_load_tile_2d(X + (size_t)m0 * IN_DIM + k1,
                                 lds_off_of(&As[buf ^ 1][0]),
                                 IN_DIM, BATCH, BK, BM, IN_DIM, 4, 3);
                tdm_load_tile_2d(W + (size_t)k1 * OUT_DIM + n0,
                                 lds_off_of(&Bs[buf ^ 1][0]),
                                 OUT_DIM, IN_DIM, BN, BK, OUT_DIM, 5, 3);
                // In-order completion: <=2 outstanding => chunk c's pair done.
                __builtin_amdgcn_s_wait_tensorcnt(2);
            } else {
                __builtin_amdgcn_s_wait_tensorcnt(0);
            }
        }
        __syncthreads();   // chunk c tiles visible to all waves

        // --- 8 x V_WMMA_F32_16X16X4_F32 over this K-chunk ---
#pragma unroll
        for (int kk = 0; kk < BK / 4; ++kk) {
            const int kb = kk * 4 + 2 * half;
            // A 16x4: v0 = K{0|2}, v1 = K{1|3}, halves split at lane 16
            v2f a;
            a.x = As[buf][arow * AS_STRIDE + kb + 0];
            a.y = As[buf][arow * AS_STRIDE + kb + 1];
            // B 4x16: row striped across lanes, same half split
            v2f b;
            b.x = Bs[buf][(kb + 0) * BS_STRIDE + bcol];
            b.y = Bs[buf][(kb + 1) * BS_STRIDE + bcol];
            acc = __builtin_amdgcn_wmma_f32_16x16x4_f32(
                false, a, false, b, (short)0, acc, false, false);
        }
        __syncthreads();   // all waves done reading buf before TDM overwrites it
    }

    // --- epilogue: C/D layout — VGPR vg holds M = vg + 8*half, N = lane&15 ---
    const int ocol = n0 + waveN * 16 + l;
    const float bv = bias[ocol];
#pragma unroll
    for (int vg = 0; vg < 8; ++vg) {
        const int orow = m0 + waveM * 16 + vg + 8 * half;
        Out[(size_t)orow * OUT_DIM + ocol] = acc[vg] + bv;
    }
}

// ---------------------------------------------------------------------------
extern "C" void kernel_launch(void* const* d_in, const int* in_sizes, int n_in,
                              void* d_out, int out_size, void* d_ws, size_t ws_size,
                              hipStream_t stream) {
    const float* x     = (const float*)d_in[0];
    const float* omega = (const float*)d_in[1];
    const float* phi   = (const float*)d_in[2];
    const float* alpha = (const float*)d_in[3];
    const float* bias  = (const float*)d_in[4];
    // d_in[5] = pos : recomputed analytically in gen_w_kernel (exact match)

    float* W = (float*)d_ws;   // 1024*1024*4 = 4 MB scratch

    gen_w_kernel<<<(IN_DIM * OUT_DIM) / 256, 256, 0, stream>>>(omega, phi, alpha, W);

    dim3 grid(OUT_DIM / BN, BATCH / BM);
    wmma_gemm_kernel<<<grid, 256, 0, stream>>>(x, W, bias, (float*)d_out);
}